// LipReadingGNN_10522669875754
// MI455X (gfx1250) — compile-verified
//
#include <hip/hip_runtime.h>

typedef __attribute__((ext_vector_type(2))) float v2f;
typedef __attribute__((ext_vector_type(8))) float v8f;

#define NG 64    // number of graphs
#define HD 128   // LSTM hidden
#define CD 500   // classes
#define FCK 256  // fc input dim

// ---------------------------------------------------------------- utilities
__global__ void fill_kernel(float* __restrict__ p, float v, int n) {
  int i = blockIdx.x * blockDim.x + threadIdx.x;
  if (i < n) p[i] = v;
}

__global__ void degree_kernel(const int* __restrict__ dst, float* __restrict__ deg, int E) {
  int e = blockIdx.x * blockDim.x + threadIdx.x;
  if (e < E) atomicAdd(&deg[dst[e]], 1.0f);
}

__global__ void invdeg_kernel(const float* __restrict__ deg, float* __restrict__ dinv,
                              float* __restrict__ sn, int N) {
  int i = blockIdx.x * blockDim.x + threadIdx.x;
  if (i < N) { float d = deg[i]; dinv[i] = rsqrtf(d); sn[i] = 1.0f / d; }
}

__global__ void enorm_kernel(const int* __restrict__ src, const int* __restrict__ dst,
                             const float* __restrict__ dinv, float* __restrict__ en, int E) {
  int e = blockIdx.x * blockDim.x + threadIdx.x;
  if (e < E) en[e] = dinv[src[e]] * dinv[dst[e]];
}

// ------------------------------------------------------------- WMMA f32 GEMM
// C(MxN) = A(MxK) @ B(KxN).  BT==true: B given as (NxK) row-major (i.e. B^T).
// One wave computes a 16x64 C tile via V_WMMA_F32_16X16X4_F32, K-step = 4,
// software-pipelined: next k-step's 5 fragment loads are issued before the
// current step's 4-WMMA chain so loads overlap the matrix pipe.
template <bool BT>
__device__ __forceinline__ void load_frags(const float* __restrict__ Arow,
                                           const float* __restrict__ B,
                                           int k, int ko, int c0, int K, int N,
                                           v2f& a, v2f& b0, v2f& b1, v2f& b2, v2f& b3) {
  a.x = Arow[k + ko];
  a.y = Arow[k + ko + 1];
  if (BT) {
    const float* B0 = B + (size_t)(c0 +  0) * K + k + ko;
    const float* B1 = B + (size_t)(c0 + 16) * K + k + ko;
    const float* B2 = B + (size_t)(c0 + 32) * K + k + ko;
    const float* B3 = B + (size_t)(c0 + 48) * K + k + ko;
    b0.x = B0[0]; b0.y = B0[1];
    b1.x = B1[0]; b1.y = B1[1];
    b2.x = B2[0]; b2.y = B2[1];
    b3.x = B3[0]; b3.y = B3[1];
  } else {
    const float* Br0 = B + (size_t)(k + ko) * N;
    const float* Br1 = B + (size_t)(k + ko + 1) * N;
    b0.x = Br0[c0 +  0]; b0.y = Br1[c0 +  0];
    b1.x = Br0[c0 + 16]; b1.y = Br1[c0 + 16];
    b2.x = Br0[c0 + 32]; b2.y = Br1[c0 + 32];
    b3.x = Br0[c0 + 48]; b3.y = Br1[c0 + 48];
  }
}

template <bool BT>
__global__ void wmma_gemm_kernel(const float* __restrict__ A, const float* __restrict__ B,
                                 float* __restrict__ C, int M, int N, int K) {
  const int lane = threadIdx.x & 31;
  const int wave = (blockIdx.x * blockDim.x + threadIdx.x) >> 5;
  const int ntiles = N >> 6;   // 64-wide tiles
  const int mtiles = M >> 4;
  if (wave >= mtiles * ntiles) return;  // wave-uniform: EXEC all-ones inside
  const int m0 = (wave / ntiles) << 4;
  const int n0 = (wave % ntiles) << 6;

  const int mrow = m0 + (lane & 15);
  const int ko   = (lane >> 4) << 1;   // 0 or 2
  const int c0   = n0 + (lane & 15);

  v8f acc0 = {}, acc1 = {}, acc2 = {}, acc3 = {};
  const float* Arow = A + (size_t)mrow * K;

  v2f a, b0, b1, b2, b3;
  load_frags<BT>(Arow, B, 0, ko, c0, K, N, a, b0, b1, b2, b3);

  for (int k = 4; k < K; k += 4) {
    v2f an, b0n, b1n, b2n, b3n;
    load_frags<BT>(Arow, B, k, ko, c0, K, N, an, b0n, b1n, b2n, b3n);
    acc0 = __builtin_amdgcn_wmma_f32_16x16x4_f32(false, a, false, b0, (short)0, acc0, false, false);
    acc1 = __builtin_amdgcn_wmma_f32_16x16x4_f32(false, a, false, b1, (short)0, acc1, false, false);
    acc2 = __builtin_amdgcn_wmma_f32_16x16x4_f32(false, a, false, b2, (short)0, acc2, false, false);
    acc3 = __builtin_amdgcn_wmma_f32_16x16x4_f32(false, a, false, b3, (short)0, acc3, false, false);
    a = an; b0 = b0n; b1 = b1n; b2 = b2n; b3 = b3n;
  }
  acc0 = __builtin_amdgcn_wmma_f32_16x16x4_f32(false, a, false, b0, (short)0, acc0, false, false);
  acc1 = __builtin_amdgcn_wmma_f32_16x16x4_f32(false, a, false, b1, (short)0, acc1, false, false);
  acc2 = __builtin_amdgcn_wmma_f32_16x16x4_f32(false, a, false, b2, (short)0, acc2, false, false);
  acc3 = __builtin_amdgcn_wmma_f32_16x16x4_f32(false, a, false, b3, (short)0, acc3, false, false);

  const int rbase = m0 + ((lane >> 4) << 3);
  for (int r = 0; r < 8; ++r) {
    float* Crow = C + (size_t)(rbase + r) * N + c0;
    Crow[0]  = acc0[r];
    Crow[16] = acc1[r];
    Crow[32] = acc2[r];
    Crow[48] = acc3[r];
  }
}

// ----------------------------------------------------------- GCN edge scatter
// agg[dst] += xw[src] * norm(e), float4 per thread; atomics land in L2
// (layer-3 agg buffer = 102 MB < 192 MB L2).
__global__ void edge_scatter_kernel(const int* __restrict__ src, const int* __restrict__ dst,
                                    const float* __restrict__ en, const float* __restrict__ X,
                                    float* __restrict__ Agg, int E, int F, int cl2) {
  int idx = blockIdx.x * blockDim.x + threadIdx.x;
  int chunks = 1 << cl2;
  if (idx >= E * chunks) return;
  int e = idx >> cl2;
  int f = (idx & (chunks - 1)) << 2;
  int s = src[e], d = dst[e];
  float nw = en[e];
  const float4 v = *(const float4*)(X + (size_t)s * F + f);
  float* o = Agg + (size_t)d * F + f;
  atomicAdd(o + 0, v.x * nw);
  atomicAdd(o + 1, v.y * nw);
  atomicAdd(o + 2, v.z * nw);
  atomicAdd(o + 3, v.w * nw);
}

// out = relu(agg + xw * self_norm + bias); Out may alias Agg (elementwise).
__global__ void combine_relu_kernel(const float* __restrict__ Agg, const float* __restrict__ G,
                                    const float* __restrict__ sn, const float* __restrict__ bias,
                                    float* Out, int N, int fl2) {
  int idx = blockIdx.x * blockDim.x + threadIdx.x;
  if (idx >= (N << fl2)) return;
  int n = idx >> fl2;
  int f = idx & ((1 << fl2) - 1);
  float v = Agg[idx] + G[idx] * sn[n] + bias[f];
  Out[idx] = v > 0.0f ? v : 0.0f;
}

// --------------------------------------------------------------- mean pooling
__global__ void pool_sum_kernel(const float* __restrict__ H, const int* __restrict__ batch,
                                float* __restrict__ sums, float* __restrict__ cnt, int N) {
  int idx = blockIdx.x * blockDim.x + threadIdx.x;  // N * 64 threads (F=256/4)
  if (idx >= N * 64) return;
  int n = idx >> 6;
  int f = (idx & 63) << 2;
  int b = batch[n];
  const float4 v = *(const float4*)(H + (size_t)n * 256 + f);
  float* o = sums + b * 256 + f;
  atomicAdd(o + 0, v.x); atomicAdd(o + 1, v.y);
  atomicAdd(o + 2, v.z); atomicAdd(o + 3, v.w);
  if (f == 0) atomicAdd(&cnt[b], 1.0f);
}

__global__ void pooled_kernel(const float* __restrict__ sums, const float* __restrict__ cnt,
                              float* __restrict__ pooled) {
  int idx = blockIdx.x * blockDim.x + threadIdx.x;
  if (idx >= NG * 256) return;
  int b = idx >> 8;
  pooled[idx] = sums[idx] / fmaxf(cnt[b], 1.0f);
}

// ------------------------------------------------- LSTM gates (T=1, h0=c0=0)
__device__ __forceinline__ float sigmoidf_(float x) { return 1.0f / (1.0f + expf(-x)); }

__global__ void lstm_gates_kernel(const float* __restrict__ gf, const float* __restrict__ gb,
                                  const float* __restrict__ bf, const float* __restrict__ bb,
                                  float* __restrict__ lout) {
  int idx = blockIdx.x * blockDim.x + threadIdx.x;
  if (idx >= NG * HD) return;
  int g = idx >> 7;
  int j = idx & (HD - 1);
  {
    const float* r = gf + g * 4 * HD;
    float i_ = r[j]          + bf[j];
    float cg = r[2 * HD + j] + bf[2 * HD + j];
    float o_ = r[3 * HD + j] + bf[3 * HD + j];
    float c  = sigmoidf_(i_) * tanhf(cg);        // forget*c_prev == 0
    lout[g * 2 * HD + j] = sigmoidf_(o_) * tanhf(c);
  }
  {
    const float* r = gb + g * 4 * HD;
    float i_ = r[j]          + bb[j];
    float cg = r[2 * HD + j] + bb[2 * HD + j];
    float o_ = r[3 * HD + j] + bb[3 * HD + j];
    float c  = sigmoidf_(i_) * tanhf(cg);
    lout[g * 2 * HD + HD + j] = sigmoidf_(o_) * tanhf(c);
  }
}

// ------------------------------------------------------------ FC + logsoftmax
__global__ void fc_logsoftmax_kernel(const float* __restrict__ lout, const float* __restrict__ fw,
                                     const float* __restrict__ fb, float* __restrict__ out) {
  __shared__ float logits[512];
  __shared__ float red[512];
  const int g = blockIdx.x;
  const int t = threadIdx.x;  // 512 threads
  float val = -3.4e38f;
  if (t < CD) {
    float acc = fb[t];
    const float* xr = lout + g * FCK;
    for (int k = 0; k < FCK; ++k) acc = fmaf(xr[k], fw[k * CD + t], acc);
    logits[t] = acc; val = acc;
  }
  red[t] = val;
  __syncthreads();
  for (int s = 256; s > 0; s >>= 1) {
    if (t < s) red[t] = fmaxf(red[t], red[t + s]);
    __syncthreads();
  }
  const float m = red[0];
  __syncthreads();
  red[t] = (t < CD) ? expf(logits[t] - m) : 0.0f;
  __syncthreads();
  for (int s = 256; s > 0; s >>= 1) {
    if (t < s) red[t] += red[t + s];
    __syncthreads();
  }
  const float lse = logf(red[0]);
  if (t < CD) out[g * CD + t] = logits[t] - m - lse;
}

// --------------------------------------------------------------------- host
static inline int cdiv_ll(long long a, int b) { return (int)((a + b - 1) / b); }

static void launch_gemm(const float* A, const float* B, float* C,
                        int M, int N, int K, bool bt, hipStream_t s) {
  int waves = (M >> 4) * (N >> 6);
  int blocks = cdiv_ll((long long)waves * 32, 256);
  if (bt) wmma_gemm_kernel<true><<<blocks, 256, 0, s>>>(A, B, C, M, N, K);
  else    wmma_gemm_kernel<false><<<blocks, 256, 0, s>>>(A, B, C, M, N, K);
}

extern "C" void kernel_launch(void* const* d_in, const int* in_sizes, int n_in,
                              void* d_out, int out_size, void* d_ws, size_t ws_size,
                              hipStream_t stream) {
  const float* x      = (const float*)d_in[0];
  const int*   eidx   = (const int*)d_in[1];
  const int*   batch  = (const int*)d_in[2];
  const float* W1     = (const float*)d_in[3];
  const float* b1     = (const float*)d_in[4];
  const float* W2     = (const float*)d_in[5];
  const float* b2     = (const float*)d_in[6];
  const float* W3     = (const float*)d_in[7];
  const float* b3     = (const float*)d_in[8];
  const float* w_ih_f = (const float*)d_in[9];
  const float* bias_f = (const float*)d_in[11];
  const float* w_ih_b = (const float*)d_in[12];
  const float* bias_b = (const float*)d_in[14];
  const float* fc_w   = (const float*)d_in[15];
  const float* fc_b   = (const float*)d_in[16];
  float* out = (float*)d_out;

  const int N = in_sizes[2];
  const int E = in_sizes[1] / 2;
  const int* esrc = eidx;
  const int* edst = eidx + E;

  // workspace carve-out (floats, 256B aligned)
  float* ws = (float*)d_ws;
  size_t off = 0;
  auto alloc = [&](size_t n) { float* p = ws + off; off += (n + 63) & ~(size_t)63; return p; };
  float* deg    = alloc(N);
  float* dinv   = alloc(N);
  float* sn     = alloc(N);
  float* en     = alloc(E);
  float* G      = alloc((size_t)N * 256);  // GEMM output per layer
  float* Agg    = alloc((size_t)N * 256);  // scatter target / final h3
  float* Hbuf   = alloc((size_t)N * 128);  // h1 (N x 64), then h2 (N x 128)
  float* sums   = alloc(NG * 256);
  float* cnt    = alloc(NG);
  float* pooled = alloc(NG * 256);
  float* gf     = alloc(NG * 512);
  float* gb     = alloc(NG * 512);
  float* lout   = alloc(NG * 256);
  (void)ws_size; (void)n_in; (void)out_size;

  // degree / normalization
  fill_kernel<<<cdiv_ll(N, 256), 256, 0, stream>>>(deg, 1.0f, N);
  degree_kernel<<<cdiv_ll(E, 256), 256, 0, stream>>>(edst, deg, E);
  invdeg_kernel<<<cdiv_ll(N, 256), 256, 0, stream>>>(deg, dinv, sn, N);
  enorm_kernel<<<cdiv_ll(E, 256), 256, 0, stream>>>(esrc, edst, dinv, en, E);

  auto gcn_layer = [&](const float* Hin, int Kf, const float* W, const float* bias,
                       int Fout, float* Hout) {
    launch_gemm(Hin, W, G, N, Fout, Kf, false, stream);                 // G = Hin @ W
    fill_kernel<<<cdiv_ll((long long)N * Fout, 256), 256, 0, stream>>>(Agg, 0.0f, N * Fout);
    int cl2 = __builtin_ctz(Fout >> 2);
    edge_scatter_kernel<<<cdiv_ll((long long)E << cl2, 256), 256, 0, stream>>>(
        esrc, edst, en, G, Agg, E, Fout, cl2);
    int fl2 = __builtin_ctz(Fout);
    combine_relu_kernel<<<cdiv_ll((long long)N * Fout, 256), 256, 0, stream>>>(
        Agg, G, sn, bias, Hout, N, fl2);
  };

  gcn_layer(x,    128, W1, b1,  64, Hbuf);  // h1
  gcn_layer(Hbuf,  64, W2, b2, 128, Hbuf);  // h2 (h1 dead after its GEMM)
  gcn_layer(Hbuf, 128, W3, b3, 256, Agg);   // h3 written in place into Agg

  // mean pool per graph
  fill_kernel<<<cdiv_ll(NG * 256, 256), 256, 0, stream>>>(sums, 0.0f, NG * 256);
  fill_kernel<<<1, 64, 0, stream>>>(cnt, 0.0f, NG);
  pool_sum_kernel<<<cdiv_ll((long long)N * 64, 256), 256, 0, stream>>>(Agg, batch, sums, cnt, N);
  pooled_kernel<<<cdiv_ll(NG * 256, 256), 256, 0, stream>>>(sums, cnt, pooled);

  // single-step bidirectional LSTM: gates = pooled @ w_ih^T  (h_prev == 0)
  launch_gemm(pooled, w_ih_f, gf, NG, 512, 256, true, stream);
  launch_gemm(pooled, w_ih_b, gb, NG, 512, 256, true, stream);
  lstm_gates_kernel<<<cdiv_ll(NG * HD, 256), 256, 0, stream>>>(gf, gb, bias_f, bias_b, lout);

  // FC + log_softmax
  fc_logsoftmax_kernel<<<NG, 512, 0, stream>>>(lout, fc_w, fc_b, out);
}